// PackedAvgPool1d_91207925497905
// MI455X (gfx1250) — compile-verified
//
#include <hip/hip_runtime.h>

// Packed avg-pool1d, K=2/S=2 fast path. Memory-bound: ~340 MB traffic, ~15 us
// at 23.3 TB/s. One wave32 per output row; 128-bit coalesced accesses via
// CDNA5 async global->LDS DMA + non-temporal stores.

#define DIM    768
#define LANES  32
#define CHUNKS (DIM / 4)        // 192 float4 chunks per row
#define CPL    (CHUNKS / LANES) // 6 chunks per lane
#define WPB    8                // waves per block
#define BLOCK  (WPB * LANES)    // 256 threads

typedef __attribute__((ext_vector_type(4))) float f32x4;

#if defined(__AMDGCN__) && __has_builtin(__builtin_amdgcn_global_load_async_to_lds_b128)
#define HAVE_ASYNC 1
// Builtin signature (from hipcc diagnostic): params are pointers to
// int __attribute__((vector_size(16))) in AS1 (global) / AS3 (LDS).
typedef int v4i __attribute__((vector_size(16)));
typedef __attribute__((address_space(1))) v4i* gv4i_t;
typedef __attribute__((address_space(3))) v4i* lv4i_t;
#else
#define HAVE_ASYNC 0
#endif

__device__ __forceinline__ void wait_async0() {
#if __has_builtin(__builtin_amdgcn_s_wait_asynccnt)
    __builtin_amdgcn_s_wait_asynccnt(0);
#else
    asm volatile("s_wait_asynccnt 0" ::: "memory");
#endif
    asm volatile("" ::: "memory"); // compiler fence: LDS reads stay after wait
}

__global__ __launch_bounds__(BLOCK) void PackedAvgPool1d_91207925497905_kernel(
    const float* __restrict__ x,
    const int*   __restrict__ gather_base,
    const int*   __restrict__ rem,
    const int*   __restrict__ ksz,
    float*       __restrict__ out,
    int Ty, int Tx)
{
    const int K    = ksz[0];
    const int lane = threadIdx.x & (LANES - 1);
    const int wib  = threadIdx.x / LANES;           // wave in block
    const int gw   = blockIdx.x * WPB + wib;        // global wave id
    const int nw   = gridDim.x * WPB;               // total waves

#if HAVE_ASYNC
    __shared__ float lds[WPB][2 * DIM];             // 6 KB per wave slot
#endif

    if (K == 2) {
        for (int row = gw; row < Ty; row += nw) {
            const int   base  = gather_base[row];
            const int   r     = rem[row];
            const int   i1    = (base + 1 < Tx) ? (base + 1) : (Tx - 1); // safe clip
            const float w1    = (r > 1) ? 1.0f : 0.0f;                   // mask 2nd tap
            const float scale = (r > 1) ? 0.5f : 1.0f;                   // 1/count
            f32x4* dst = (f32x4*)(out + (size_t)row * DIM);

#if HAVE_ASYNC
            // WAR guard: previous iteration's LDS reads must retire before the
            // async DMA writes land in the same slot.
            asm volatile("s_wait_dscnt 0" ::: "memory");

            gv4i_t g0 = (gv4i_t)(void*)(x + (size_t)base * DIM) + lane;
            gv4i_t g1 = (gv4i_t)(void*)(x + (size_t)i1   * DIM) + lane;
            lv4i_t l0 = (lv4i_t)(void*)&lds[wib][0] + lane;
            lv4i_t l1 = l0 + CHUNKS;   // second row slot, +3072 bytes

            // 6 x 512B-strided b128 async copies per source row (imm offset
            // applies to both global and LDS address per ISA 08 §4.4).
            __builtin_amdgcn_global_load_async_to_lds_b128(g0, l0,    0, 0);
            __builtin_amdgcn_global_load_async_to_lds_b128(g0, l0,  512, 0);
            __builtin_amdgcn_global_load_async_to_lds_b128(g0, l0, 1024, 0);
            __builtin_amdgcn_global_load_async_to_lds_b128(g0, l0, 1536, 0);
            __builtin_amdgcn_global_load_async_to_lds_b128(g0, l0, 2048, 0);
            __builtin_amdgcn_global_load_async_to_lds_b128(g0, l0, 2560, 0);
            __builtin_amdgcn_global_load_async_to_lds_b128(g1, l1,    0, 0);
            __builtin_amdgcn_global_load_async_to_lds_b128(g1, l1,  512, 0);
            __builtin_amdgcn_global_load_async_to_lds_b128(g1, l1, 1024, 0);
            __builtin_amdgcn_global_load_async_to_lds_b128(g1, l1, 1536, 0);
            __builtin_amdgcn_global_load_async_to_lds_b128(g1, l1, 2048, 0);
            __builtin_amdgcn_global_load_async_to_lds_b128(g1, l1, 2560, 0);

            wait_async0();

            const f32x4* s = (const f32x4*)&lds[wib][0];
            #pragma unroll
            for (int c = 0; c < CPL; ++c) {
                const int idx = lane + c * LANES;
                f32x4 a = s[idx];
                f32x4 b = s[CHUNKS + idx];
                f32x4 o = (a + b * w1) * scale;
                __builtin_nontemporal_store(o, dst + idx); // write-once stream
            }
#else
            const f32x4* s0 = (const f32x4*)(x + (size_t)base * DIM);
            const f32x4* s1 = (const f32x4*)(x + (size_t)i1   * DIM);
            #pragma unroll
            for (int c = 0; c < CPL; ++c) {
                const int idx = lane + c * LANES;
                f32x4 a = s0[idx];
                f32x4 b = s1[idx];
                f32x4 o = (a + b * w1) * scale;
                __builtin_nontemporal_store(o, dst + idx);
            }
#endif
        }
    } else {
        // Generic K fallback (reference semantics: mask i < rem, /min(rem,K)).
        for (int row = gw; row < Ty; row += nw) {
            const int   base  = gather_base[row];
            const int   r     = rem[row];
            const int   cnt   = (r < K) ? r : K;   // taps actually summed
            const float scale = 1.0f / (float)cnt;
            f32x4 acc[CPL];
            #pragma unroll
            for (int c = 0; c < CPL; ++c) acc[c] = f32x4{0.f, 0.f, 0.f, 0.f};
            for (int k = 0; k < cnt; ++k) {
                int ik = base + k; if (ik > Tx - 1) ik = Tx - 1;
                const f32x4* s = (const f32x4*)(x + (size_t)ik * DIM);
                #pragma unroll
                for (int c = 0; c < CPL; ++c) acc[c] += s[lane + c * LANES];
            }
            f32x4* dst = (f32x4*)(out + (size_t)row * DIM);
            #pragma unroll
            for (int c = 0; c < CPL; ++c)
                __builtin_nontemporal_store(acc[c] * scale, dst + lane + c * LANES);
        }
    }
}

extern "C" void kernel_launch(void* const* d_in, const int* in_sizes, int n_in,
                              void* d_out, int out_size, void* d_ws, size_t ws_size,
                              hipStream_t stream) {
    (void)n_in; (void)d_ws; (void)ws_size;
    const float* x           = (const float*)d_in[0];
    const int*   gather_base = (const int*)  d_in[1];
    const int*   rem         = (const int*)  d_in[2];
    const int*   ksz         = (const int*)  d_in[3];
    float*       out         = (float*)d_out;

    const int Ty = in_sizes[1];          // one output row per gather_base entry
    const int Tx = in_sizes[0] / DIM;    // packed input rows
    (void)out_size;                      // == Ty * DIM

    int grid = (Ty + WPB - 1) / WPB;     // one wave per row, 8 waves per block
    if (grid < 1) grid = 1;

    PackedAvgPool1d_91207925497905_kernel<<<grid, BLOCK, 0, stream>>>(
        x, gather_base, rem, ksz, out, Ty, Tx);
}